// QuestAttention_15135464751582
// MI455X (gfx1250) — compile-verified
//
#include <hip/hip_runtime.h>
#include <hip/hip_bf16.h>
#include <math.h>

#define HID   4096
#define NH    32
#define HD    128
#define PAGE  16
#define NPAGE 1024
#define TOPK  128

typedef float v2f __attribute__((ext_vector_type(2)));
typedef float v8f __attribute__((ext_vector_type(8)));

// ---------------------------------------------------------------------------
// 16-output GEMV tile via V_WMMA_F32_16X16X4_F32.
// A (16x4) has only row 0 live (= x chunk); B (4x16) holds Wᵀ chunk.
// A layout: lanes 0-15 = M rows, vgpr v = K=v (lanes<16) / K=v+2 (lanes>=16)
// B layout: lanes 0-15 = N cols, vgpr v = K=v (lanes<16) / K=v+2 (lanes>=16)
// => lane L loads the contiguous float2 W[r0+(L&15)][k + 2*(L>>4) .. +1].
// D row 0 (out) lands in acc[0], lanes 0..15.
// ---------------------------------------------------------------------------
__device__ inline void gemv16_wmma(const float* __restrict__ W,
                                   const float* __restrict__ x,
                                   float* __restrict__ out, int r0) {
  const int lane = threadIdx.x & 31;
  const int lo   = lane & 15;
  const int hi   = lane >> 4;                 // 0 or 1 -> K offset 0 / 2
  const float* wrow = W + (size_t)(r0 + lo) * HID + hi * 2;
  const float* xptr = x + hi * 2;
  const bool live = (lo == 0);
  v2f z; z[0] = 0.f; z[1] = 0.f;
  v8f acc0 = {}; v8f acc1 = {};
  for (int k = 0; k < HID; k += 8) {
    if ((k & 31) == 0) __builtin_prefetch(wrow + k + 512, 0, 1);
    v2f b0 = *(const v2f*)(wrow + k);
    v2f b1 = *(const v2f*)(wrow + k + 4);
    v2f a0r = *(const v2f*)(xptr + k);
    v2f a1r = *(const v2f*)(xptr + k + 4);
    v2f a0 = live ? a0r : z;
    v2f a1 = live ? a1r : z;
    acc0 = __builtin_amdgcn_wmma_f32_16x16x4_f32(false, a0, false, b0,
                                                 (short)0, acc0, false, false);
    acc1 = __builtin_amdgcn_wmma_f32_16x16x4_f32(false, a1, false, b1,
                                                 (short)0, acc1, false, false);
  }
  if (lane < 16) out[r0 + lane] = acc0[0] + acc1[0];
}

// K1: q,k,v = h @ W{q,k,v}ᵀ.  768 tiles, 8 waves/block -> 96 blocks.
__global__ void __launch_bounds__(256)
qkv_wmma_kernel(const float* __restrict__ Wq, const float* __restrict__ Wk,
                const float* __restrict__ Wv, const float* __restrict__ h,
                float* __restrict__ qkv) {
  int tile = blockIdx.x * 8 + (threadIdx.x >> 5);   // 0..767
  int m    = tile >> 8;                             // 0=q 1=k 2=v
  int r0   = (tile & 255) * 16;
  const float* W = (m == 0) ? Wq : (m == 1) ? Wk : Wv;
  gemv16_wmma(W, h, qkv + m * HID, r0);
}

// K2: RoPE in place on q (blocks 0..31) and k (blocks 32..63).
__global__ void rope_kernel(float* __restrict__ qk, const int* __restrict__ posp) {
  int b = blockIdx.x;           // 64 blocks
  int j = threadIdx.x;          // 0..63  (half-dim)
  float* x = qk + ((b < NH) ? (b * HD) : (HID + (b - NH) * HD));
  float pos = (float)(*posp);
  float inv = powf(10000.0f, -(float)j / 64.0f);
  float c, s;
  sincosf(pos * inv, &s, &c);
  float x1 = x[j], x2 = x[j + 64];
  x[j]      = x1 * c - x2 * s;
  x[j + 64] = x2 * c + x1 * s;
}

// K3: per (page<1023, head) min/max over 16 slots + Quest score. 1 wave each.
__global__ void __launch_bounds__(256)
scores_kernel(const float* __restrict__ kc, const float* __restrict__ q,
              float* __restrict__ scores) {
  int wid  = (blockIdx.x * blockDim.x + threadIdx.x) >> 5;
  int lane = threadIdx.x & 31;
  int p = wid >> 5;
  int h = wid & 31;
  if (p >= NPAGE - 1) return;
  const float* base = kc + (((size_t)p * PAGE) * NH + h) * HD + lane * 4;
  float4 mn = *(const float4*)base;
  float4 mx = mn;
  for (int e = 1; e < PAGE; ++e) {
    float4 v = *(const float4*)(base + (size_t)e * NH * HD);
    mn.x = fminf(mn.x, v.x); mx.x = fmaxf(mx.x, v.x);
    mn.y = fminf(mn.y, v.y); mx.y = fmaxf(mx.y, v.y);
    mn.z = fminf(mn.z, v.z); mx.z = fmaxf(mx.z, v.z);
    mn.w = fminf(mn.w, v.w); mx.w = fmaxf(mx.w, v.w);
  }
  float4 qv = *(const float4*)(q + h * HD + lane * 4);
  float s = fmaxf(qv.x * mn.x, qv.x * mx.x) + fmaxf(qv.y * mn.y, qv.y * mx.y)
          + fmaxf(qv.z * mn.z, qv.z * mx.z) + fmaxf(qv.w * mn.w, qv.w * mx.w);
  for (int off = 16; off > 0; off >>= 1) s += __shfl_xor(s, off, 32);
  if (lane == 0) scores[h * NPAGE + p] = s;
}

// K4: top-128 per head by iterative argmax in LDS. One block per head.
__global__ void __launch_bounds__(256)
topk_kernel(const float* __restrict__ scores, int* __restrict__ idx) {
  __shared__ float sc[NPAGE];
  __shared__ float rv[256];
  __shared__ int   ri[256];
  int h = blockIdx.x, t = threadIdx.x;
  for (int i = t; i < NPAGE; i += 256)
    sc[i] = (i < NPAGE - 1) ? scores[h * NPAGE + i] : -__builtin_inff();
  __syncthreads();
  for (int sel = 0; sel < TOPK; ++sel) {
    float bv = -__builtin_inff(); int bi = 0;
    for (int i = t; i < NPAGE; i += 256)
      if (sc[i] > bv) { bv = sc[i]; bi = i; }
    rv[t] = bv; ri[t] = bi;
    __syncthreads();
    for (int off = 128; off > 0; off >>= 1) {
      if (t < off && rv[t + off] > rv[t]) { rv[t] = rv[t + off]; ri[t] = ri[t + off]; }
      __syncthreads();
    }
    if (t == 0) { idx[h * 132 + sel] = ri[0]; sc[ri[0]] = -__builtin_inff(); }
    __syncthreads();
  }
  if (t == 0) idx[h * 132 + TOPK] = NPAGE - 1;
}

// K5: gathered attention, one block per head, 8 waves with online softmax.
// (page 1023, slot 15) is substituted with the fresh RoPE'd k/v — the cache
// is never written (inputs must stay immutable).
__global__ void __launch_bounds__(256)
attn_kernel(const float* __restrict__ kc, const float* __restrict__ vc,
            const float* __restrict__ q,  const float* __restrict__ knew,
            const float* __restrict__ vnew, const int* __restrict__ idx,
            float* __restrict__ attn) {
  int h = blockIdx.x;
  int w = threadIdx.x >> 5, lane = threadIdx.x & 31;
  float4 qv = *(const float4*)(q + h * HD + lane * 4);
  float m = -__builtin_inff(), s = 0.f;
  float4 vacc = {0.f, 0.f, 0.f, 0.f};
  for (int slot = w; slot < TOPK + 1; slot += 8) {
    int page = idx[h * 132 + slot];
    for (int e = 0; e < PAGE; ++e) {
      bool isnew = (page == NPAGE - 1) && (e == PAGE - 1);
      size_t off = (((size_t)page * PAGE + e) * NH + h) * HD;
      const float* kp = isnew ? (knew + h * HD) : (kc + off);
      const float* vp = isnew ? (vnew + h * HD) : (vc + off);
      float4 kv = *(const float4*)(kp + lane * 4);
      float d = qv.x * kv.x + qv.y * kv.y + qv.z * kv.z + qv.w * kv.w;
      for (int o = 16; o > 0; o >>= 1) d += __shfl_xor(d, o, 32);
      float logit = d * 0.08838834764831845f;   // 1/sqrt(128)
      float mn = fmaxf(m, logit);
      float sc = __expf(m - mn);
      float wg = __expf(logit - mn);
      float4 vv = *(const float4*)(vp + lane * 4);
      vacc.x = vacc.x * sc + wg * vv.x;
      vacc.y = vacc.y * sc + wg * vv.y;
      vacc.z = vacc.z * sc + wg * vv.z;
      vacc.w = vacc.w * sc + wg * vv.w;
      s = s * sc + wg;
      m = mn;
    }
  }
  __shared__ float lm[8], ls[8], lv[8][HD];
  __shared__ float gM, gS;
  if (lane == 0) { lm[w] = m; ls[w] = s; }
  lv[w][lane * 4 + 0] = vacc.x;
  lv[w][lane * 4 + 1] = vacc.y;
  lv[w][lane * 4 + 2] = vacc.z;
  lv[w][lane * 4 + 3] = vacc.w;
  __syncthreads();
  if (threadIdx.x == 0) {
    float mm = lm[0];
    for (int i = 1; i < 8; ++i) mm = fmaxf(mm, lm[i]);
    float ss = 0.f;
    for (int i = 0; i < 8; ++i) ss += ls[i] * __expf(lm[i] - mm);
    gM = mm; gS = ss;
  }
  __syncthreads();
  if (threadIdx.x < HD) {
    int d = threadIdx.x;
    float a = 0.f;
    for (int i = 0; i < 8; ++i) a += __expf(lm[i] - gM) * lv[i][d];
    attn[h * HD + d] = a / gS;
  }
}

// K6: out = attn @ Woᵀ.  256 tiles -> 32 blocks × 8 waves.
__global__ void __launch_bounds__(256)
out_wmma_kernel(const float* __restrict__ Wo, const float* __restrict__ attn,
                float* __restrict__ out) {
  int tile = blockIdx.x * 8 + (threadIdx.x >> 5);
  gemv16_wmma(Wo, attn, out, tile * 16);
}

extern "C" void kernel_launch(void* const* d_in, const int* in_sizes, int n_in,
                              void* d_out, int out_size, void* d_ws, size_t ws_size,
                              hipStream_t stream) {
  (void)in_sizes; (void)n_in; (void)out_size; (void)ws_size;
  const float* h   = (const float*)d_in[0];
  const float* Wq  = (const float*)d_in[1];
  const float* Wk  = (const float*)d_in[2];
  const float* Wv  = (const float*)d_in[3];
  const float* Wo  = (const float*)d_in[4];
  const float* kc  = (const float*)d_in[5];
  const float* vc  = (const float*)d_in[6];
  const int*   pos = (const int*)d_in[7];
  float* out = (float*)d_out;

  // workspace layout (floats)
  float* ws     = (float*)d_ws;
  float* qkv    = ws;                    // q[4096] k[4096] v[4096]
  float* qv     = ws;                    // q after rope (in place)
  float* knew   = ws + HID;              // k after rope (in place)
  float* vnew   = ws + 2 * HID;          // v (raw)
  float* scores = ws + 3 * HID;          // 32 x 1024
  int*   idx    = (int*)(ws + 3 * HID + NH * NPAGE);   // 32 x 132
  float* attn   = ws + 3 * HID + NH * NPAGE + (NH * 132); // 4096

  qkv_wmma_kernel<<<96, 256, 0, stream>>>(Wq, Wk, Wv, h, qkv);
  rope_kernel<<<64, 64, 0, stream>>>(qkv, pos);
  scores_kernel<<<4092, 256, 0, stream>>>(kc, qv, scores);
  topk_kernel<<<NH, 256, 0, stream>>>(scores, idx);
  attn_kernel<<<NH, 256, 0, stream>>>(kc, vc, qv, knew, vnew, idx, attn);
  out_wmma_kernel<<<32, 256, 0, stream>>>(Wo, attn, out);
}